// DeepGCN_38543036514657
// MI455X (gfx1250) — compile-verified
//
#include <hip/hip_runtime.h>
#include <hip/hip_bf16.h>

#define N_NODES 50000
#define N_EDGES 800000
#define H 64
#define L 4
#define F_NODE 128
#define F_EDGE 8
#define C_OUT 112
#define EPS 1e-7f
#define LN_EPS 1e-5f
#define SM_EPS 1e-16f

typedef __attribute__((ext_vector_type(2))) float v2f;
typedef __attribute__((ext_vector_type(8))) float v8f;

// ---------------------------------------------------------------------------
// monotone float <-> uint encoding so atomicMax(uint) == float max (all signs)
// ---------------------------------------------------------------------------
__device__ __forceinline__ unsigned fenc(float f) {
    unsigned u = __float_as_uint(f);
    return (u & 0x80000000u) ? ~u : (u | 0x80000000u);
}
__device__ __forceinline__ float fdec(unsigned u) {
    return (u & 0x80000000u) ? __uint_as_float(u & 0x7FFFFFFFu)
                             : __uint_as_float(~u);
}

__device__ __forceinline__ v8f wmma_f32_k4(v2f a, v2f b, v8f c) {
    // V_WMMA_F32_16X16X4_F32: full-precision f32 matrix op (wave32)
    return __builtin_amdgcn_wmma_f32_16x16x4_f32(
        false, a, false, b, (short)0, c, false, false);
}

// ---------------------------------------------------------------------------
// Generic C[rows x M] = A[rows x K] @ W[K x M] + bias ; rows%16==0, M%16==0,
// K%4==0. blockDim = (M/16)*32, gridDim.x = rows/16. One 16x16 tile per wave.
// ---------------------------------------------------------------------------
__global__ void __launch_bounds__(256)
wmma_gemm_bias(const float* __restrict__ A, const float* __restrict__ W,
               const float* __restrict__ bias, float* __restrict__ C,
               int K, int M) {
    const int wave = threadIdx.x >> 5;
    const int lane = threadIdx.x & 31;
    const int half = lane >> 4;        // K-pair select
    const int ln   = lane & 15;        // M index for A, N index for B
    const int row0 = blockIdx.x << 4;
    const int colg = (wave << 4) + ln;

    v8f acc = {};
    const float* Arow = A + (size_t)(row0 + ln) * K;
    for (int k = 0; k < K; k += 4) {
        const int ko = k + 2 * half;
        v2f a; a.x = Arow[ko];            a.y = Arow[ko + 1];
        v2f b; b.x = W[(size_t)ko * M + colg]; b.y = W[(size_t)(ko + 1) * M + colg];
        acc = wmma_f32_k4(a, b, acc);
    }
    const float bb = bias[colg];
#pragma unroll
    for (int rr = 0; rr < 8; ++rr) {
        const int row = row0 + rr + 8 * half;
        C[(size_t)row * M + colg] = acc[rr] + bb;
    }
}

// ---------------------------------------------------------------------------
// h = relu(layernorm(x, g, b))  over rows of width H=64. One wave per row.
// ---------------------------------------------------------------------------
__global__ void __launch_bounds__(256)
prenorm_relu(const float* __restrict__ x, const float* __restrict__ g,
             const float* __restrict__ b, float* __restrict__ out, int nrows) {
    const int wave = threadIdx.x >> 5;
    const int lane = threadIdx.x & 31;
    const int row  = blockIdx.x * 8 + wave;
    if (row >= nrows) return;
    const float v0 = x[(size_t)row * H + lane];
    const float v1 = x[(size_t)row * H + 32 + lane];
    float s = v0 + v1;
    float q = v0 * v0 + v1 * v1;
#pragma unroll
    for (int off = 16; off > 0; off >>= 1) {
        s += __shfl_xor(s, off, 32);
        q += __shfl_xor(q, off, 32);
    }
    const float mu = s * (1.0f / 64.0f);
    const float rs = rsqrtf(q * (1.0f / 64.0f) - mu * mu + LN_EPS);
    out[(size_t)row * H + lane]      = fmaxf((v0 - mu) * rs * g[lane]      + b[lane],      0.0f);
    out[(size_t)row * H + 32 + lane] = fmaxf((v1 - mu) * rs * g[lane + 32] + b[lane + 32], 0.0f);
}

// ---------------------------------------------------------------------------
// per-layer reset of softmax accumulators
// ---------------------------------------------------------------------------
__global__ void init_layer(unsigned* __restrict__ menc, float* __restrict__ num,
                           float* __restrict__ den, int n) {
    const int i = blockIdx.x * 256 + threadIdx.x;
    if (i < n) { menc[i] = 0u; num[i] = 0.0f; den[i] = 0.0f; }
}

// ---------------------------------------------------------------------------
// Edge pass 1: msg = relu(x[src]+ea)+eps ; atomic-max of logits per (dst,ch).
// ea recomputed on the fly from edge_attr (8-dot vs We in LDS) -> saves HBM.
// 256 threads = 4 edges x 64 channels.
// ---------------------------------------------------------------------------
__global__ void __launch_bounds__(256)
edge_max_pass(const int* __restrict__ src, const int* __restrict__ dst,
              const float* __restrict__ eattr, const float* __restrict__ We,
              const float* __restrict__ be, const float* __restrict__ xnode,
              const float* __restrict__ tptr, int layer,
              unsigned* __restrict__ menc, int nedges) {
    __shared__ float WeS[F_EDGE * H];
    __shared__ float beS[H];
    __shared__ float att[4][F_EDGE];
    const int tid = threadIdx.x;
    for (int i = tid; i < F_EDGE * H; i += 256) WeS[i] = We[i];
    if (tid < H) beS[tid] = be[tid];
    const int el = tid >> 6;           // edge within block
    const int c  = tid & 63;           // channel
    const int e  = blockIdx.x * 4 + el;
    if (c < F_EDGE && e < nedges) att[el][c] = eattr[(size_t)e * F_EDGE + c];
    __syncthreads();
    if (e >= nedges) return;
    const int s = src[e], d = dst[e];
    float ea = beS[c];
#pragma unroll
    for (int k = 0; k < F_EDGE; ++k) ea += att[el][k] * WeS[k * H + c];
    const float msg   = fmaxf(xnode[(size_t)s * H + c] + ea, 0.0f) + EPS;
    const float logit = msg * tptr[layer];
    atomicMax(&menc[(size_t)d * H + c], fenc(logit));
}

// ---------------------------------------------------------------------------
// Edge pass 2: e = exp(logit - max); den += e ; num += msg*e  (f32 atomics,
// accumulator tables live in L2). out = num/(den+eps) later.
// ---------------------------------------------------------------------------
__global__ void __launch_bounds__(256)
edge_sum_pass(const int* __restrict__ src, const int* __restrict__ dst,
              const float* __restrict__ eattr, const float* __restrict__ We,
              const float* __restrict__ be, const float* __restrict__ xnode,
              const float* __restrict__ tptr, int layer,
              const unsigned* __restrict__ menc, float* __restrict__ num,
              float* __restrict__ den, int nedges) {
    __shared__ float WeS[F_EDGE * H];
    __shared__ float beS[H];
    __shared__ float att[4][F_EDGE];
    const int tid = threadIdx.x;
    for (int i = tid; i < F_EDGE * H; i += 256) WeS[i] = We[i];
    if (tid < H) beS[tid] = be[tid];
    const int el = tid >> 6;
    const int c  = tid & 63;
    const int e  = blockIdx.x * 4 + el;
    if (c < F_EDGE && e < nedges) att[el][c] = eattr[(size_t)e * F_EDGE + c];
    __syncthreads();
    if (e >= nedges) return;
    const int s = src[e], d = dst[e];
    float ea = beS[c];
#pragma unroll
    for (int k = 0; k < F_EDGE; ++k) ea += att[el][k] * WeS[k * H + c];
    const float msg   = fmaxf(xnode[(size_t)s * H + c] + ea, 0.0f) + EPS;
    const float tv    = tptr[layer];
    const float m     = fdec(menc[(size_t)d * H + c]);
    const float ex    = __expf(msg * tv - m);
    atomicAdd(&den[(size_t)d * H + c], ex);
    atomicAdd(&num[(size_t)d * H + c], msg * ex);
}

// ---------------------------------------------------------------------------
// Fused node update for one GENConv layer, 16 nodes per block, 8 waves:
//   out = num/(den+sm_eps) + xin                       (root residual)
//   t1  = out @ W1 + b1          (WMMA f32, 8 waves x 16x16 tiles, K=64)
//   t1  = relu(LN(t1,g1,c1))     (block-wide over width 128)
//   y   = t1 @ W2 + b2           (WMMA f32, 4 waves, K=128)
//   xout = y (+ xres if res+ block)
// ---------------------------------------------------------------------------
#define APAD 68
#define HPAD 132

__global__ void __launch_bounds__(256)
node_conv_mlp(const float* __restrict__ num, const float* __restrict__ den,
              const float* __restrict__ xin, const float* __restrict__ xres,
              float* __restrict__ xout,
              const float* __restrict__ W1, const float* __restrict__ b1,
              const float* __restrict__ g1, const float* __restrict__ c1,
              const float* __restrict__ W2, const float* __restrict__ b2,
              int with_res) {
    __shared__ float Alds[16 * APAD];      // 16 x 64 "out" tile
    __shared__ float Hlds[16 * HPAD];      // 16 x 128 hidden tile
    __shared__ float redS[16][17], redQ[16][17];
    __shared__ float muS[16], rsS[16];

    const int tid  = threadIdx.x;
    const int row0 = blockIdx.x << 4;
    const int wave = tid >> 5;
    const int lane = tid & 31;
    const int half = lane >> 4;
    const int ln   = lane & 15;

    // 1) root residual + softmax-aggregate finalize -> LDS
    for (int i = tid; i < 16 * H; i += 256) {
        const int r = i >> 6, c = i & 63;
        const size_t g = (size_t)(row0 + r) * H + c;
        Alds[r * APAD + c] = num[g] / (den[g] + SM_EPS) + xin[g];
    }
    __syncthreads();

    // 2) GEMM1: [16x64] @ [64x128] ; wave w -> columns [16w,16w+16)
    {
        const int colg = (wave << 4) + ln;
        v8f acc = {};
        for (int k = 0; k < H; k += 4) {
            const int ko = k + 2 * half;
            v2f a; a.x = Alds[ln * APAD + ko]; a.y = Alds[ln * APAD + ko + 1];
            v2f b; b.x = W1[ko * 128 + colg];  b.y = W1[(ko + 1) * 128 + colg];
            acc = wmma_f32_k4(a, b, acc);
        }
        const float bb = b1[colg];
#pragma unroll
        for (int rr = 0; rr < 8; ++rr)
            Hlds[(rr + 8 * half) * HPAD + colg] = acc[rr] + bb;
    }
    __syncthreads();

    // 3) LayerNorm(width 128) + ReLU, 16 threads per row
    {
        const int row = tid >> 4, seg = tid & 15;
        float s = 0.0f, q = 0.0f;
#pragma unroll
        for (int j = 0; j < 8; ++j) {
            const float v = Hlds[row * HPAD + seg * 8 + j];
            s += v; q += v * v;
        }
        redS[row][seg] = s; redQ[row][seg] = q;
    }
    __syncthreads();
    if (tid < 16) {
        float s = 0.0f, q = 0.0f;
#pragma unroll
        for (int j = 0; j < 16; ++j) { s += redS[tid][j]; q += redQ[tid][j]; }
        const float mu = s * (1.0f / 128.0f);
        muS[tid] = mu;
        rsS[tid] = rsqrtf(q * (1.0f / 128.0f) - mu * mu + LN_EPS);
    }
    __syncthreads();
    {
        const int row = tid >> 4, seg = tid & 15;
        const float mu = muS[row], rs = rsS[row];
#pragma unroll
        for (int j = 0; j < 8; ++j) {
            const int c = seg * 8 + j;
            const float v = (Hlds[row * HPAD + c] - mu) * rs * g1[c] + c1[c];
            Hlds[row * HPAD + c] = fmaxf(v, 0.0f);
        }
    }
    __syncthreads();

    // 4) GEMM2: [16x128] @ [128x64] ; waves 0-3 (uniform branch, EXEC full)
    if (wave < 4) {
        const int colg = (wave << 4) + ln;
        v8f acc = {};
        for (int k = 0; k < 128; k += 4) {
            const int ko = k + 2 * half;
            v2f a; a.x = Hlds[ln * HPAD + ko]; a.y = Hlds[ln * HPAD + ko + 1];
            v2f b; b.x = W2[ko * H + colg];    b.y = W2[(ko + 1) * H + colg];
            acc = wmma_f32_k4(a, b, acc);
        }
        const float bb = b2[colg];
#pragma unroll
        for (int rr = 0; rr < 8; ++rr) {
            const int row = rr + 8 * half;
            const size_t g = (size_t)(row0 + row) * H + colg;
            float y = acc[rr] + bb;
            if (with_res) y += xres[g];
            xout[g] = y;
        }
    }
}

// ---------------------------------------------------------------------------
extern "C" void kernel_launch(void* const* d_in, const int* in_sizes, int n_in,
                              void* d_out, int out_size, void* d_ws, size_t ws_size,
                              hipStream_t stream) {
    const float* x     = (const float*)d_in[0];
    const int*   ei    = (const int*)  d_in[1];   // [2,E]
    const float* eattr = (const float*)d_in[2];
    const float* Wn    = (const float*)d_in[3];
    const float* bn    = (const float*)d_in[4];
    const float* We    = (const float*)d_in[5];
    const float* be    = (const float*)d_in[6];
    const float* t     = (const float*)d_in[7];
    const float* W1    = (const float*)d_in[8];   // [L,64,128]
    const float* b1    = (const float*)d_in[9];   // [L,128]
    const float* g1    = (const float*)d_in[10];
    const float* c1    = (const float*)d_in[11];
    const float* W2    = (const float*)d_in[12];  // [L,128,64]
    const float* b2    = (const float*)d_in[13];  // [L,64]
    const float* gL    = (const float*)d_in[14];  // [L,64]
    const float* bL    = (const float*)d_in[15];
    const float* Wo    = (const float*)d_in[16];  // [64,112]
    const float* bo    = (const float*)d_in[17];

    const int* src = ei;
    const int* dst = ei + N_EDGES;

    const size_t NH = (size_t)N_NODES * H;
    float*    xbuf = (float*)d_ws;
    float*    hbuf = xbuf + NH;
    unsigned* menc = (unsigned*)(hbuf + NH);
    float*    numb = (float*)(menc + NH);
    float*    denb = numb + NH;

    const int rowBlocks  = N_NODES / 16;            // 3125
    const int edgeBlocks = (N_EDGES + 3) / 4;       // 200000
    const int initBlocks = (int)((NH + 255) / 256);

    // node encoder: x[N,128] @ Wn[128,64] + bn  -> xbuf
    wmma_gemm_bias<<<rowBlocks, 128, 0, stream>>>(x, Wn, bn, xbuf, F_NODE, H);

    for (int i = 0; i < L; ++i) {
        const float* xin = xbuf;
        if (i > 0) {
            prenorm_relu<<<(N_NODES + 7) / 8, 256, 0, stream>>>(
                xbuf, gL + i * H, bL + i * H, hbuf, N_NODES);
            xin = hbuf;
        }
        init_layer<<<initBlocks, 256, 0, stream>>>(menc, numb, denb, (int)NH);
        edge_max_pass<<<edgeBlocks, 256, 0, stream>>>(
            src, dst, eattr, We, be, xin, t, i, menc, N_EDGES);
        edge_sum_pass<<<edgeBlocks, 256, 0, stream>>>(
            src, dst, eattr, We, be, xin, t, i, menc, numb, denb, N_EDGES);
        node_conv_mlp<<<rowBlocks, 256, 0, stream>>>(
            numb, denb, xin, xbuf, xbuf,
            W1 + (size_t)i * H * 128, b1 + (size_t)i * 128,
            g1 + (size_t)i * 128,     c1 + (size_t)i * 128,
            W2 + (size_t)i * 128 * H, b2 + (size_t)i * H,
            i > 0 ? 1 : 0);
    }

    // head: relu(LN(x, gL[0], bL[0])) @ Wo[64,112] + bo
    prenorm_relu<<<(N_NODES + 7) / 8, 256, 0, stream>>>(
        xbuf, gL, bL, hbuf, N_NODES);
    wmma_gemm_bias<<<rowBlocks, 224, 0, stream>>>(
        hbuf, Wo, bo, (float*)d_out, H, C_OUT);
}